// GCNBlock_16260746182822
// MI455X (gfx1250) — compile-verified
//
#include <hip/hip_runtime.h>

#define N_NODES  40000
#define N_EDGES  640000
#define PK   64            // weight panel height (K rows per stage)
#define ESTR 132           // e1 tile LDS row stride (132 % 64 == 4 -> half-disjoint banks)
#define HSTR 132           // hidden tile LDS row stride

typedef float v2f __attribute__((ext_vector_type(2)));
typedef float v8f __attribute__((ext_vector_type(8)));

__device__ __forceinline__ v8f wmma_f32(v2f a, v2f b, v8f c) {
  // D(16x16 f32) = A(16x4 f32) x B(4x16 f32) + C
  return __builtin_amdgcn_wmma_f32_16x16x4_f32(
      /*neg_a=*/false, a, /*neg_b=*/false, b,
      /*c_mod=*/(short)0, c, /*reuse_a=*/false, /*reuse_b=*/false);
}

__device__ __forceinline__ void atomic_add_f32(float* p, float v) {
  unsafeAtomicAdd(p, v);   // global_atomic_add_f32
}

__device__ __forceinline__ uint32_t lds_off(const void* p) {
  // flat LDS aperture: addr[31:0] is the LDS byte offset
  return (uint32_t)(uintptr_t)p;
}

// Async-copy one contiguous 64x128 f32 weight panel (32 KB) global -> LDS.
// 256 threads x 8 x b128; tracked on ASYNCcnt (no VGPR round-trip).
__device__ __forceinline__ void stage_panel_async(float* ldsdst, const float* gsrc) {
  const int tid = threadIdx.x;
#pragma unroll
  for (int it = 0; it < 8; ++it) {
    const int fi = (it * 256 + tid) * 4;
    asm volatile("global_load_async_to_lds_b128 %0, %1, off"
                 :: "v"(lds_off(ldsdst + fi)), "v"(gsrc + fi)
                 : "memory");
  }
}

__device__ __forceinline__ void wait_async0() {
  asm volatile("s_wait_asynccnt 0x0" ::: "memory");
}

// 64-row (4 M-tiles), 2-layer MLP: out = relu(A @ W1 + b1) @ W2 + b2.
// 8 waves: wave w -> M-tile (w>>1), column half (w&1) as 4 N-tiles of 16.
// A columns come in 128-wide segments: segment s from per-lane row pointer
// pa_s (global, row stride 128), or from the LDS ebuf tile when s == EBUFSEG.
// W1/W2 are PRE-REPACKED pair-interleaved: Wp[((k>>1)*128 + n)*2 + (k&1)],
// so a B fragment is one aligned 8-byte LDS load into an adjacent VGPR pair.
// Panels (64 K-rows = 32 pair-rows, contiguous 32 KB) are async-staged, 2-buffered.
template<int KDIM, int EBUFSEG, class Epi>
__device__ __forceinline__ void mlp64(
    const float* pa0, const float* pa1, const float* pa2,
    const float* ebuf,
    const float* __restrict__ W1, const float* __restrict__ B1,
    const float* __restrict__ W2, const float* __restrict__ B2,
    float* wpan /* [2][PK*128] */, float* hid /* [64*HSTR] */,
    Epi epi)
{
  const int tid  = threadIdx.x;
  const int wave = tid >> 5, lane = tid & 31;
  const int half = lane >> 4, lm = lane & 15;
  const int mt   = wave >> 1, nh = wave & 1;
  const int r    = mt * 16 + lm;      // A row this lane feeds
  const int ncb  = nh * 64 + lm;      // column base for B/D (+ nt*16)

  constexpr int NP1 = KDIM / PK;

  // ---- GEMM1: hid = relu(A @ W1 + b1) ----
  stage_panel_async(wpan, W1);
  wait_async0();
  __syncthreads();

  v8f acc[4];
#pragma unroll
  for (int nt = 0; nt < 4; ++nt) acc[nt] = (v8f){0.f,0.f,0.f,0.f,0.f,0.f,0.f,0.f};

#pragma unroll 1
  for (int p = 0; p < NP1; ++p) {
    if (p + 1 < NP1)  // overlap DMA of next panel with WMMA on current one
      stage_panel_async(wpan + ((p + 1) & 1) * (PK * 128), W1 + (size_t)(p + 1) * (PK * 128));
    const float* wp = wpan + (p & 1) * (PK * 128);
#pragma unroll
    for (int kk = 0; kk < PK; kk += 4) {
      const int k   = p * PK + kk;
      const int s   = k >> 7;                 // uniform: input segment
      const int kka = (k & 127) + 2 * half;   // A col pair (16x4 f32 layout)
      v2f a;
      if (s == EBUFSEG) {
        const float2 t = *(const float2*)&ebuf[r * ESTR + kka];
        a[0] = t.x; a[1] = t.y;
      } else {
        const float* pa = (s == 0) ? pa0 : (s == 1) ? pa1 : pa2;
        const float2 t = *(const float2*)(pa + kka);
        a[0] = t.x; a[1] = t.y;
      }
      const int kp = (kk >> 1) + half;        // panel-local K pair index
      const float* wpp = wp + (kp << 8);      // kp*256
#pragma unroll
      for (int nt = 0; nt < 4; ++nt) {
        const int nc = ncb + nt * 16;
        v2f b = *(const v2f*)(wpp + nc * 2);  // {W[kw][nc], W[kw+1][nc]}
        acc[nt] = wmma_f32(a, b, acc[nt]);
      }
    }
    wait_async0();
    __syncthreads();
  }

  // bias + relu -> hid (C/D layout: row = g + 8*half within M-tile)
#pragma unroll
  for (int nt = 0; nt < 4; ++nt) {
    const float bb = B1[ncb + nt * 16];
#pragma unroll
    for (int g = 0; g < 8; ++g) {
      float v = acc[nt][g] + bb;
      hid[(mt * 16 + g + 8 * half) * HSTR + ncb + nt * 16] = v > 0.f ? v : 0.f;
    }
  }
  __syncthreads();

  // ---- GEMM2: out = hid @ W2 + b2 ----
  stage_panel_async(wpan, W2);
  wait_async0();
  __syncthreads();

  v8f acc2[4];
#pragma unroll
  for (int nt = 0; nt < 4; ++nt) acc2[nt] = (v8f){0.f,0.f,0.f,0.f,0.f,0.f,0.f,0.f};

#pragma unroll 1
  for (int p = 0; p < 2; ++p) {
    if (p == 0) stage_panel_async(wpan + PK * 128, W2 + PK * 128);
    const float* wp = wpan + (p & 1) * (PK * 128);
#pragma unroll
    for (int kk = 0; kk < PK; kk += 4) {
      const int kka = p * PK + kk + 2 * half;
      const float2 t = *(const float2*)&hid[r * HSTR + kka];
      v2f a; a[0] = t.x; a[1] = t.y;
      const int kp = (kk >> 1) + half;
      const float* wpp = wp + (kp << 8);
#pragma unroll
      for (int nt = 0; nt < 4; ++nt) {
        const int nc = ncb + nt * 16;
        v2f b = *(const v2f*)(wpp + nc * 2);
        acc2[nt] = wmma_f32(a, b, acc2[nt]);
      }
    }
    wait_async0();
    __syncthreads();
  }

#pragma unroll
  for (int nt = 0; nt < 4; ++nt) {
    const float bb = B2[ncb + nt * 16];
#pragma unroll
    for (int g = 0; g < 8; ++g)
      epi(mt * 16 + g + 8 * half, ncb + nt * 16, acc2[nt][g] + bb);
  }
}

__global__ __launch_bounds__(256) void k_zero(float* __restrict__ p, size_t n) {
  size_t i = (size_t)blockIdx.x * 256 + threadIdx.x;
  if (i < n) p[i] = 0.f;
}

// Pair-interleave repack: Wp[((k>>1)*128 + n)*2 + (k&1)] = W[k*128 + n]
__global__ __launch_bounds__(256) void k_repack(const float* __restrict__ W,
                                                float* __restrict__ Wp, int kdim) {
  const int i = blockIdx.x * 256 + threadIdx.x;
  if (i < kdim * 128) {
    const int k = i >> 7, n = i & 127;
    Wp[(((k >> 1) << 7) + n) * 2 + (k & 1)] = W[i];
  }
}

// ---- Kernel 2: e1 = MLP([x[row]|x[col]|ea]); msg = MLP([x[row]|e1]); x1 += scatter(msg, col)
__global__ __launch_bounds__(256) void k_e1_msg(
    const float* __restrict__ x, const float* __restrict__ ea, const int* __restrict__ eidx,
    const float* __restrict__ e1w1, const float* __restrict__ e1b1,
    const float* __restrict__ e1w2, const float* __restrict__ e1b2,
    const float* __restrict__ n1w1, const float* __restrict__ n1b1,
    const float* __restrict__ n1w2, const float* __restrict__ n1b2,
    float* __restrict__ e1_out, float* __restrict__ x1)
{
  __shared__ float wpan[2 * PK * 128];
  __shared__ float hid[64 * HSTR];
  __shared__ float ebuf[64 * ESTR];
  __shared__ int rowi[64], coli[64];

  const int tid = threadIdx.x;
  const size_t e0 = (size_t)blockIdx.x * 64;
  if (tid < 64) {
    rowi[tid] = eidx[e0 + tid];
    coli[tid] = eidx[(size_t)N_EDGES + e0 + tid];
  }
  __syncthreads();

  const int wave = tid >> 5, lane = tid & 31;
  const int mt = wave >> 1, lm = lane & 15;
  const int r = mt * 16 + lm;
  const float* pa0 = x + (size_t)rowi[r] * 128;
  const float* pa1 = x + (size_t)coli[r] * 128;
  const float* pa2 = ea + (e0 + (size_t)r) * 128;

  float* e1p = e1_out + e0 * 128;
  mlp64<384, -1>(pa0, pa1, pa2, nullptr, e1w1, e1b1, e1w2, e1b2, wpan, hid,
      [&](int row, int col, float v) {
        e1p[row * 128 + col] = v;        // e1 to scratch (used by kernels 3/4)
        ebuf[row * ESTR + col] = v;      // e1 tile for the fused msg MLP
      });
  __syncthreads();

  mlp64<256, 1>(pa0, nullptr, nullptr, ebuf, n1w1, n1b1, n1w2, n1b2, wpan, hid,
      [&](int row, int col, float v) {
        atomic_add_f32(&x1[(size_t)coli[row] * 128 + col], v);
      });
}

// ---- Kernel 3: msg2 = MLP([x1[row]|e1]); x2 += scatter(msg2, col)
__global__ __launch_bounds__(256) void k_msg2(
    const float* __restrict__ x1, const float* __restrict__ e1g, const int* __restrict__ eidx,
    const float* __restrict__ n2w1, const float* __restrict__ n2b1,
    const float* __restrict__ n2w2, const float* __restrict__ n2b2,
    float* __restrict__ x2)
{
  __shared__ float wpan[2 * PK * 128];
  __shared__ float hid[64 * HSTR];
  __shared__ int rowi[64], coli[64];

  const int tid = threadIdx.x;
  const size_t e0 = (size_t)blockIdx.x * 64;
  if (tid < 64) {
    rowi[tid] = eidx[e0 + tid];
    coli[tid] = eidx[(size_t)N_EDGES + e0 + tid];
  }
  __syncthreads();

  const int wave = tid >> 5, lane = tid & 31;
  const int mt = wave >> 1, lm = lane & 15;
  const int r = mt * 16 + lm;
  const float* pa0 = x1 + (size_t)rowi[r] * 128;
  const float* pa1 = e1g + (e0 + (size_t)r) * 128;

  mlp64<256, -1>(pa0, pa1, nullptr, nullptr, n2w1, n2b1, n2w2, n2b2, wpan, hid,
      [&](int row, int col, float v) {
        atomic_add_f32(&x2[(size_t)coli[row] * 128 + col], v);
      });
}

// ---- Kernel 4: e2 = MLP([x2[row]|x2[col]|e1])  (all three segments global)
__global__ __launch_bounds__(256) void k_e2(
    const float* __restrict__ x2g, const float* __restrict__ e1g, const int* __restrict__ eidx,
    const float* __restrict__ e2w1, const float* __restrict__ e2b1,
    const float* __restrict__ e2w2, const float* __restrict__ e2b2,
    float* __restrict__ e2_out)
{
  __shared__ float wpan[2 * PK * 128];
  __shared__ float hid[64 * HSTR];
  __shared__ int rowi[64], coli[64];

  const int tid = threadIdx.x;
  const size_t e0 = (size_t)blockIdx.x * 64;
  if (tid < 64) {
    rowi[tid] = eidx[e0 + tid];
    coli[tid] = eidx[(size_t)N_EDGES + e0 + tid];
  }
  __syncthreads();

  const int wave = tid >> 5, lane = tid & 31;
  const int mt = wave >> 1, lm = lane & 15;
  const int r = mt * 16 + lm;
  const float* pa0 = x2g + (size_t)rowi[r] * 128;
  const float* pa1 = x2g + (size_t)coli[r] * 128;
  const float* pa2 = e1g + (e0 + (size_t)r) * 128;

  float* e2p = e2_out + e0 * 128;
  mlp64<384, -1>(pa0, pa1, pa2, nullptr, e2w1, e2b1, e2w2, e2b2, wpan, hid,
      [&](int row, int col, float v) {
        e2p[row * 128 + col] = v;
      });
}

extern "C" void kernel_launch(void* const* d_in, const int* in_sizes, int n_in,
                              void* d_out, int out_size, void* d_ws, size_t ws_size,
                              hipStream_t stream) {
  const float* x    = (const float*)d_in[0];
  const float* ea   = (const float*)d_in[1];
  const int*   eidx = (const int*)d_in[2];
  const float* e1w1 = (const float*)d_in[3];
  const float* e1b1 = (const float*)d_in[4];
  const float* e1w2 = (const float*)d_in[5];
  const float* e1b2 = (const float*)d_in[6];
  const float* n1w1 = (const float*)d_in[7];
  const float* n1b1 = (const float*)d_in[8];
  const float* n1w2 = (const float*)d_in[9];
  const float* n1b2 = (const float*)d_in[10];
  const float* n2w1 = (const float*)d_in[11];
  const float* n2b1 = (const float*)d_in[12];
  const float* n2w2 = (const float*)d_in[13];
  const float* n2b2 = (const float*)d_in[14];
  const float* e2w1 = (const float*)d_in[15];
  const float* e2b1 = (const float*)d_in[16];
  const float* e2w2 = (const float*)d_in[17];
  const float* e2b2 = (const float*)d_in[18];

  float* out = (float*)d_out;
  float* x2  = out;                              // N_NODES x 128
  float* e2  = out + (size_t)N_NODES * 128;      // N_EDGES x 128

  // workspace layout
  float* x1  = (float*)d_ws;                     // N_NODES x 128
  float* e1  = x1 + (size_t)N_NODES * 128;       // N_EDGES x 128
  float* wsp = e1 + (size_t)N_EDGES * 128;       // repacked weights
  float* p_e1w1 = wsp;                 // 384*128
  float* p_e1w2 = p_e1w1 + 384 * 128;  // 128*128
  float* p_n1w1 = p_e1w2 + 128 * 128;  // 256*128
  float* p_n1w2 = p_n1w1 + 256 * 128;  // 128*128
  float* p_n2w1 = p_n1w2 + 128 * 128;  // 256*128
  float* p_n2w2 = p_n2w1 + 256 * 128;  // 128*128
  float* p_e2w1 = p_n2w2 + 128 * 128;  // 384*128
  float* p_e2w2 = p_e2w1 + 384 * 128;  // 128*128

  // one-time (per launch) pair-interleave weight repack (~917 KB total)
  k_repack<<<(384 * 128 + 255) / 256, 256, 0, stream>>>(e1w1, p_e1w1, 384);
  k_repack<<<(128 * 128 + 255) / 256, 256, 0, stream>>>(e1w2, p_e1w2, 128);
  k_repack<<<(256 * 128 + 255) / 256, 256, 0, stream>>>(n1w1, p_n1w1, 256);
  k_repack<<<(128 * 128 + 255) / 256, 256, 0, stream>>>(n1w2, p_n1w2, 128);
  k_repack<<<(256 * 128 + 255) / 256, 256, 0, stream>>>(n2w1, p_n2w1, 256);
  k_repack<<<(128 * 128 + 255) / 256, 256, 0, stream>>>(n2w2, p_n2w2, 128);
  k_repack<<<(384 * 128 + 255) / 256, 256, 0, stream>>>(e2w1, p_e2w1, 384);
  k_repack<<<(128 * 128 + 255) / 256, 256, 0, stream>>>(e2w2, p_e2w2, 128);

  const size_t nz = (size_t)N_NODES * 128;
  k_zero<<<(unsigned)((nz + 255) / 256), 256, 0, stream>>>(x1, nz);
  k_zero<<<(unsigned)((nz + 255) / 256), 256, 0, stream>>>(x2, nz);

  const unsigned nblk = N_EDGES / 64;   // 10000, exact
  k_e1_msg<<<nblk, 256, 0, stream>>>(x, ea, eidx,
                                     p_e1w1, e1b1, p_e1w2, e1b2,
                                     p_n1w1, n1b1, p_n1w2, n1b2,
                                     e1, x1);
  k_msg2<<<nblk, 256, 0, stream>>>(x1, e1, eidx,
                                   p_n2w1, n2b1, p_n2w2, n2b2,
                                   x2);
  k_e2<<<nblk, 256, 0, stream>>>(x2, e1, eidx,
                                 p_e2w1, e2b1, p_e2w2, e2b2,
                                 e2);
}